// clipRefine_69140383531137
// MI455X (gfx1250) — compile-verified
//
#include <hip/hip_runtime.h>
#include <hip/hip_bf16.h>
#include <stdint.h>

// CDNA5 wave32 WMMA types
typedef float v2f __attribute__((ext_vector_type(2)));
typedef float v8f __attribute__((ext_vector_type(8)));

#define KTOT     1024   // inner (C_IN) dimension for both GEMMs
#define NTOT     128    // positions
#define KCHUNK   64     // K rows of X staged in LDS per step
#define NCHUNK   (KTOT / KCHUNK)
#define LDSTRIDE 132    // 128 + 4 pad: (2*LDSTRIDE) % 64 != 0 -> halves hit different banks

// Y[b, m, n] = relu( sum_k W[m,k] * X[b,k,n] + bias[m] )
// Block = 128 threads (4 waves). Each wave: one 16-row m-tile x all 8 n-tiles.
// V_WMMA_F32_16X16X4_F32 (exact fp32 on the matrix pipe) + double-buffered
// GLOBAL_LOAD_ASYNC_TO_LDS_B128 staging (memory->LDS DMA, ASYNCcnt-tracked).
__global__ __launch_bounds__(128) void gemm_bias_relu_wmma(
    const float* __restrict__ W,     // (Mtot, KTOT) row-major
    const float* __restrict__ X,     // (B, KTOT, NTOT)
    const float* __restrict__ bias,  // (Mtot)
    float* __restrict__ Y,           // (B, Mtot, NTOT)
    int Mtot)
{
    __shared__ float lds[2][KCHUNK * LDSTRIDE];

    const int tid  = threadIdx.x;
    const int lane = tid & 31;
    const int wave = tid >> 5;     // 0..3
    const int hf   = lane >> 4;    // lane half: K-pair selector (0 -> K+0/1, 1 -> K+2/3)
    const int l    = lane & 15;    // M index (A frag) / N index (B frag)

    const int mgroups = Mtot / 64;            // 4 m-tiles (64 rows) per block
    const int b  = blockIdx.x / mgroups;
    const int mg = blockIdx.x % mgroups;
    const int m0 = mg * 64 + wave * 16;       // this wave's m-tile base row

    const float* Xb = X + (size_t)b * KTOT * NTOT;
    const float* Wm = W + (size_t)m0 * KTOT;

    // Async DMA stage of X[k0..k0+63][0..127] into lds[buf] (padded rows).
    // Each wave loads rows wave, wave+4, ... ; 16B per lane per row.
    // LDS dest addr = low 32 bits of the flat shared pointer (LDS aperture
    // keeps the workgroup-relative offset in addr[31:0]); 16B aligned.
    auto stage = [&](int k0, int buf) {
        #pragma unroll
        for (int r = 0; r < KCHUNK / 4; ++r) {
            const int row = wave + 4 * r;
            const float* g = Xb + (size_t)(k0 + row) * NTOT + lane * 4;
            const unsigned loff =
                (unsigned)(uintptr_t)&lds[buf][row * LDSTRIDE + lane * 4];
            asm volatile("global_load_async_to_lds_b128 %0, %1, off"
                         :: "v"(loff), "v"(g) : "memory");
        }
    };

    v8f acc[8] = {};   // 8 n-tiles x (16x16 f32 tile / 8 VGPRs)

    stage(0, 0);
    asm volatile("s_wait_asynccnt 0x0" ::: "memory");
    __syncthreads();

    for (int c = 0; c < NCHUNK; ++c) {
        // Prefetch next chunk into the other buffer while we compute.
        // Safe: the barrier ending iteration c-1 guaranteed every wave is
        // done reading lds[(c+1)&1] (it held chunk c-1).
        if (c + 1 < NCHUNK) stage((c + 1) * KCHUNK, (c + 1) & 1);

        const float* L  = lds[c & 1];
        const int    k0 = c * KCHUNK;

        #pragma unroll
        for (int kk = 0; kk < KCHUNK; kk += 4) {
            const int krow = kk + hf * 2;
            // A fragment: 16x4 f32. lane l holds W[m0+l][k+hf*2 .. +1] (contiguous float2).
            v2f a;
            {
                const float2 av = *(const float2*)(Wm + (size_t)l * KTOT + (k0 + krow));
                a.x = av.x; a.y = av.y;
            }
            // 8 back-to-back WMMAs reuse the A fragment.
            #pragma unroll
            for (int t = 0; t < 8; ++t) {
                v2f bf;
                bf.x = L[(krow + 0) * LDSTRIDE + t * 16 + l];
                bf.y = L[(krow + 1) * LDSTRIDE + t * 16 + l];
                acc[t] = __builtin_amdgcn_wmma_f32_16x16x4_f32(
                    false, a, false, bf, (short)0, acc[t], false, false);
            }
        }

        if (c + 1 < NCHUNK) {
            // My chunk-(c+1) DMAs complete, then block-wide rendezvous
            // (__syncthreads waits DScnt first -> all reads of chunk c done).
            asm volatile("s_wait_asynccnt 0x0" ::: "memory");
            __syncthreads();
        }
    }

    // Epilogue: C/D layout -> VGPR r holds row m0 + r + hf*8, col = t*16 + l.
    #pragma unroll
    for (int r = 0; r < 8; ++r) {
        const int row = m0 + r + hf * 8;
        const float bv = bias[row];
        float* Yrow = Y + ((size_t)b * Mtot + row) * NTOT;
        #pragma unroll
        for (int t = 0; t < 8; ++t) {
            float v = acc[t][r] + bv;
            Yrow[t * 16 + l] = v > 0.0f ? v : 0.0f;
        }
    }
}

// d[b,n] = (1/512) * sum_c tanh(state[b,c,n] - state[b,c,n+1]),  n in [0,127)
// (Only the super/sub-diagonal of A_sta survives the tridiagonal mask; tanh is odd.)
__global__ __launch_bounds__(512) void edge_tanh_kernel(
    const float* __restrict__ state,  // (B, 512, 128)
    float* __restrict__ d)            // (B, 128)
{
    __shared__ float part[512];
    const int b   = blockIdx.x;
    const int tid = threadIdx.x;
    const int n   = tid & 127;
    const int cc  = tid >> 7;         // channel chunk 0..3 (128 channels each)

    float s = 0.0f;
    if (n < 127) {
        const float* sp = state + ((size_t)b * 512 + (size_t)cc * 128) * 128;
        for (int c = 0; c < 128; ++c) {
            s += tanhf(sp[c * 128 + n] - sp[c * 128 + n + 1]);
        }
    }
    part[tid] = s;
    __syncthreads();
    if (tid < 128) {
        const float tot = part[n] + part[n + 128] + part[n + 256] + part[n + 384];
        d[b * 128 + n] = (n < 127) ? tot * (1.0f / 512.0f) : 0.0f;
    }
}

// h[b,c,m] = x[b,c,m] + d[b,m-1]*x[b,c,m-1] - d[b,m]*x[b,c,m+1]
// (x · A_sta collapsed to a 3-point stencil.)
__global__ __launch_bounds__(256) void aggregate_kernel(
    const float* __restrict__ x,   // (B, 1024, 128)
    const float* __restrict__ d,   // (B, 128)
    float* __restrict__ h)         // (B, 1024, 128)
{
    const int idx = blockIdx.x * 256 + threadIdx.x;   // < 8*1024*128
    const int m   = idx & 127;
    const int b   = idx >> 17;                        // idx / (1024*128)
    float v = x[idx];
    if (m > 0)   v += d[b * 128 + m - 1] * x[idx - 1];
    if (m < 127) v -= d[b * 128 + m]     * x[idx + 1];
    h[idx] = v;
}

extern "C" void kernel_launch(void* const* d_in, const int* in_sizes, int n_in,
                              void* d_out, int out_size, void* d_ws, size_t ws_size,
                              hipStream_t stream) {
    (void)in_sizes; (void)n_in; (void)out_size; (void)ws_size;
    const float* x  = (const float*)d_in[0];   // (8, 1024, 128)
    const float* W1 = (const float*)d_in[1];   // (512, 1024)
    const float* b1 = (const float*)d_in[2];   // (512)
    const float* W2 = (const float*)d_in[3];   // (1024, 1024)
    const float* b2 = (const float*)d_in[4];   // (1024)
    // d_in[5] = A (|i-j|<2 mask), d_in[6] = I : constant structure folded analytically.

    float* ws    = (float*)d_ws;
    float* state = ws;                                   // 8*512*128
    float* dcoef = state + (size_t)8 * 512 * 128;        // 8*128
    float* h     = dcoef + (size_t)8 * 128;              // 8*1024*128
    float* y     = (float*)d_out;                        // (8, 1024, 128)

    // state = relu(W1 @ x + b1)          : 64 blocks x 128 thr
    gemm_bias_relu_wmma<<<8 * (512 / 64), 128, 0, stream>>>(W1, x, b1, state, 512);
    // d[b,n] = mean_c tanh(state diff)   : 8 blocks x 512 thr
    edge_tanh_kernel<<<8, 512, 0, stream>>>(state, dcoef);
    // h = tridiagonal stencil of x        : 4096 blocks x 256 thr
    aggregate_kernel<<<(8 * 1024 * 128) / 256, 256, 0, stream>>>(x, dcoef, h);
    // y = relu(W2 @ h + b2)              : 128 blocks x 128 thr
    gemm_bias_relu_wmma<<<8 * (1024 / 64), 128, 0, stream>>>(W2, h, b2, y, 1024);
}